// GroupingUnit2D_12764642804238
// MI455X (gfx1250) — compile-verified
//
#include <hip/hip_runtime.h>
#include <hip/hip_bf16.h>

// ---------------------------------------------------------------------------
// GroupingUnit2D for MI455X (gfx1250, wave32, WMMA).
//
// Phase 1 (assign_kernel):  logits = min(2 x.Wt - |x|^2 - |c|^2, 0)/sigmoid(sf)
//                           assign = softmax_P(logits)          [N,64]
//   - W (64x512 fp32 = 128KB) staged k-pair-interleaved into LDS so every
//     B-fragment is a single aligned ds_load_b64 (no register glue).
//   - per wave: one 16-row tile, 4x v_wmma_f32_16x16x4_f32 accumulators
//     (16x64 logits), fused clamp + softmax via wave32 butterflies.
//
// Phase 2 (pool_kernel):  out[g] = assign[g]^T @ feats[g]   (K = S*B = 6144)
//   - grid (g, 32-wide c-tile); 512-thread blocks, 16 waves split K;
//     each wave: 8 accumulators (4 p-tiles x 2 c-tiles) so the 4 A-fragment
//     loads feed 8 WMMAs; cross-wave reduction via LDS ds_add_f32.
// ---------------------------------------------------------------------------

typedef __attribute__((ext_vector_type(2))) float v2f;
typedef __attribute__((ext_vector_type(8))) float v8f;

constexpr int S  = 3;
constexpr int G  = 8;
constexpr int Bn = 2048;
constexpr int C  = 512;
constexpr int P  = 64;
constexpr int N  = S * G * Bn;   // 49152 rows total
constexpr int KG = S * Bn;       // 6144 rows per graph

__device__ __forceinline__ v8f wmma_f32_16x16x4(v2f a, v2f b, v8f c) {
  // D(16x16,f32) = A(16x4,f32) * B(4x16,f32) + C
  return __builtin_amdgcn_wmma_f32_16x16x4_f32(
      /*neg_a=*/false, a, /*neg_b=*/false, b,
      /*c_mod=*/(short)0, c, /*reuse_a=*/false, /*reuse_b=*/false);
}

__device__ __forceinline__ float half_group_sum(float v) {
  v += __shfl_xor(v, 1);
  v += __shfl_xor(v, 2);
  v += __shfl_xor(v, 4);
  v += __shfl_xor(v, 8);
  return v;
}

__device__ __forceinline__ float half_group_max(float v) {
  v = fmaxf(v, __shfl_xor(v, 1));
  v = fmaxf(v, __shfl_xor(v, 2));
  v = fmaxf(v, __shfl_xor(v, 4));
  v = fmaxf(v, __shfl_xor(v, 8));
  return v;
}

// ---------------------------------------------------------------------------
// Kernel 1: fused RBF-score + softmax -> assign [N, P]
// block = 256 threads (8 waves); each wave owns one 16-row tile.
// grid = N / 128.
// ---------------------------------------------------------------------------
__global__ __launch_bounds__(256)
void assign_kernel(const float* __restrict__ x, const float* __restrict__ w,
                   const float* __restrict__ sf, float* __restrict__ assign)
{
  // Pair-interleaved transposed weight: lds_w[(k/2)*2P + 2p + (k&1)] = w[p][k]
  // -> fragment {w[p][kk], w[p][kk+1]} is one aligned float2 in LDS.
  __shared__ float lds_w[C * P];     // 128 KB ([C/2][P][2])
  __shared__ float lds_csq[P];       // |c_p|^2
  __shared__ float lds_beta[P];      // sigmoid(smooth_factor)

  const int tid = threadIdx.x;

  // Stage: each thread moves one k-pair; consecutive lanes take consecutive p
  // so the ds b64 stores are bank-conflict-free (weight is L2-resident, the
  // strided global reads only touch 128 KB).
  for (int e = tid; e < (C * P) / 2; e += 256) {
    const int p  = e & (P - 1);
    const int kp = e >> 6;           // k-pair index
    const v2f val = *reinterpret_cast<const v2f*>(w + p * C + kp * 2);
    *reinterpret_cast<v2f*>(&lds_w[kp * (2 * P) + 2 * p]) = val;
  }
  __syncthreads();

  if (tid < P) {
    float s = 0.0f;
    for (int kp = 0; kp < C / 2; ++kp) {
      const v2f v = *reinterpret_cast<const v2f*>(&lds_w[kp * (2 * P) + 2 * tid]);
      s += v.x * v.x + v.y * v.y;
    }
    lds_csq[tid]  = s;
    lds_beta[tid] = 1.0f / (1.0f + __expf(-sf[tid]));
  }
  __syncthreads();

  const int wave = tid >> 5;
  const int lane = tid & 31;
  const int half = lane >> 4;        // 0: lanes 0-15, 1: lanes 16-31
  const int l15  = lane & 15;
  const int rowBase = (blockIdx.x * 8 + wave) * 16;
  const int koff = half * 2;         // A-frag K slice: {0,1} or {2,3}

  const float* __restrict__ xrow = x + (size_t)(rowBase + l15) * C;
  // This lane's half always reads pair-row (k/2 + half).
  const float* __restrict__ wbase = lds_w + half * (2 * P) + 2 * l15;

  v8f acc[4] = {};                   // 16x64 logits (4 col tiles)
  float xsq = 0.0f;                  // partial |x_row|^2 (K parity split)

  for (int k = 0; k < C; k += 4) {
    const v2f a = *reinterpret_cast<const v2f*>(xrow + k + koff);
    xsq += a.x * a.x + a.y * a.y;
    __builtin_prefetch(xrow + k + 64, 0, 0);   // global_prefetch_b8
    const float* __restrict__ wp = wbase + (k >> 1) * (2 * P);
#pragma unroll
    for (int t = 0; t < 4; ++t) {
      const v2f b = *reinterpret_cast<const v2f*>(wp + t * 32);  // one ds_load_b64
      acc[t] = wmma_f32_16x16x4(a, b, acc[t]);
    }
  }

  // Combine the two K-parity halves: lane l and lane l^16 hold the same row.
  xsq += __shfl_xor(xsq, 16);

  float csq[4], beta[4];
#pragma unroll
  for (int t = 0; t < 4; ++t) {
    csq[t]  = lds_csq[t * 16 + l15];   // matches C/D col mapping N = lane&15
    beta[t] = lds_beta[t * 16 + l15];
  }

#pragma unroll
  for (int v = 0; v < 8; ++v) {
    const int m = v + half * 8;                 // C/D row for this VGPR
    const float xs = __shfl(xsq, m);            // |x|^2 of row m (any half)
    float lg[4];
#pragma unroll
    for (int t = 0; t < 4; ++t)
      lg[t] = fminf(2.0f * acc[t][v] - xs - csq[t], 0.0f) / beta[t];

    float mx = fmaxf(fmaxf(lg[0], lg[1]), fmaxf(lg[2], lg[3]));
    mx = half_group_max(mx);

    float e[4];
    float sum = 0.0f;
#pragma unroll
    for (int t = 0; t < 4; ++t) { e[t] = __expf(lg[t] - mx); sum += e[t]; }
    sum = half_group_sum(sum);
    const float inv = 1.0f / sum;

    const size_t o = (size_t)(rowBase + m) * P + l15;
#pragma unroll
    for (int t = 0; t < 4; ++t)
      assign[o + t * 16] = e[t] * inv;
  }
}

// ---------------------------------------------------------------------------
// Kernel 2: out[g,p,c] = sum_k assign[r(g,k), p] * x[r(g,k), c]
// grid = G * (C/32); block = 512 (16 waves split K=6144, each wave 4 p-tiles
// x 2 c-tiles = 8 WMMA accumulators); LDS f32-atomic cross-wave reduce.
// ---------------------------------------------------------------------------
__global__ __launch_bounds__(512)
void pool_kernel(const float* __restrict__ x, const float* __restrict__ assign,
                 float* __restrict__ out)
{
  __shared__ float red[P * 32];      // 64 x 32 output tile (8 KB)

  const int tid   = threadIdx.x;
  const int g     = blockIdx.x >> 4;       // 16 c-tiles (of 32) per graph
  const int cBase = (blockIdx.x & 15) * 32;

  for (int i = tid; i < P * 32; i += 512) red[i] = 0.0f;
  __syncthreads();

  const int wave = tid >> 5;         // 0..15
  const int lane = tid & 31;
  const int half = lane >> 4;
  const int l15  = lane & 15;
  const int koff = half * 2;
  const int c0   = cBase + l15;

  v8f acc[8] = {};                   // [t][j]: p-tile t, c-tile j
  const int k0 = wave * (KG / 16);   // 384 K-rows per wave

  for (int k = k0; k < k0 + KG / 16; k += 4) {
    const int kk = k + koff;         // kk is even -> rows r0, r0+1 same batch
    // flat row index r = s*(G*B) + g*B + n  for k = s*B + n
    const size_t r0 = (size_t)(kk >> 11) * (G * Bn) + (size_t)g * Bn + (kk & (Bn - 1));

    const float* __restrict__ ap = assign + r0 * P;   // rows r0 and r0+1
    const float* __restrict__ xp = x + r0 * C;

    v2f b0, b1;                      // B-frags: feats [k x c] for 2 c-tiles
    b0.x = xp[c0];
    b0.y = xp[C + c0];
    b1.x = xp[c0 + 16];
    b1.y = xp[C + c0 + 16];
    __builtin_prefetch(xp + 2 * C + c0, 0, 0);

#pragma unroll
    for (int t = 0; t < 4; ++t) {    // A-frag: assign^T [p x k]
      v2f a;
      a.x = ap[t * 16 + l15];        // assign[r0  ][p]
      a.y = ap[P + t * 16 + l15];    // assign[r0+1][p]
      acc[t * 2 + 0] = wmma_f32_16x16x4(a, b0, acc[t * 2 + 0]);
      acc[t * 2 + 1] = wmma_f32_16x16x4(a, b1, acc[t * 2 + 1]);
    }
  }

  // Cross-wave reduction: within a wave each (p,c) slot is unique, across the
  // 16 waves we need atomic ds adds.
#pragma unroll
  for (int t = 0; t < 4; ++t) {
#pragma unroll
    for (int v = 0; v < 8; ++v) {
      const int p = t * 16 + v + half * 8;
      atomicAdd(&red[p * 32 + l15],      acc[t * 2 + 0][v]);
      atomicAdd(&red[p * 32 + 16 + l15], acc[t * 2 + 1][v]);
    }
  }
  __syncthreads();

  for (int i = tid; i < P * 32; i += 512) {
    const int p  = i >> 5;
    const int cc = i & 31;
    out[((size_t)g * P + p) * C + cBase + cc] = red[i];
  }
}

// ---------------------------------------------------------------------------
extern "C" void kernel_launch(void* const* d_in, const int* in_sizes, int n_in,
                              void* d_out, int out_size, void* d_ws, size_t ws_size,
                              hipStream_t stream) {
  (void)in_sizes; (void)n_in; (void)out_size; (void)d_ws; (void)ws_size;
  const float* x  = (const float*)d_in[0];   // node_feats [S, G*B, C] fp32
  const float* w  = (const float*)d_in[1];   // weight     [P, C]      fp32
  const float* sf = (const float*)d_in[2];   // smooth_factor [P]      fp32
  // d_out = outputs [G,P,C] ++ assign [N,P], both fp32, all elements written.
  float* outputs = (float*)d_out;
  float* assign  = outputs + (size_t)G * P * C;

  assign_kernel<<<N / 128, 256, 0, stream>>>(x, w, sf, assign);
  pool_kernel<<<G * (C / 32), 512, 0, stream>>>(x, assign, outputs);
}